// PMBNet_1340029796456
// MI455X (gfx1250) — compile-verified
//
#include <hip/hip_runtime.h>
#include <cstdint>

// Problem geometry (fixed by the reference's setup_inputs).
#define N_  8
#define C_  2
#define H_  720
#define W_  1280
constexpr int HW   = H_ * W_;        // 921,600  (divisible by 4)
constexpr int NHW  = N_ * HW;        // 7,372,800 pixels
constexpr int NCHW = N_ * C_ * HW;   // 14,745,600 elems per output tensor

// ---------------------------------------------------------------------------
// Device-scope native FP32 atomic add (CDNA5 GLOBAL_ATOMIC_ADD_F32, no-return,
// STOREcnt-tracked). RMW resolves in the 192MB L2, which holds the entire
// accumulator footprint (G: 59MB + norm: 29.5MB = 88.5MB resident).
// ---------------------------------------------------------------------------
__device__ __forceinline__ void atom_add_f32_dev(float* p, float v) {
    asm volatile("global_atomic_add_f32 %0, %1, off scope:SCOPE_DEV"
                 :: "v"((uint64_t)(uintptr_t)p), "v"(v)
                 : "memory");
}

// ---------------------------------------------------------------------------
// Zero-fill (float4 -> GLOBAL_STORE_B128).
// ---------------------------------------------------------------------------
__global__ void fwarp_zero_kernel(float4* __restrict__ p, int n4) {
    int i = blockIdx.x * blockDim.x + threadIdx.x;
    if (i < n4) p[i] = make_float4(0.f, 0.f, 0.f, 0.f);
}

// ---------------------------------------------------------------------------
// Splat: one thread per source pixel. Accumulates
//   G    = (1-t)*f01w - t*f10w          (2 channels, stored in out1 half)
//   norm = (1-t)*norm0 + t*norm1        (1 plane, in d_ws)
// via 3 atomics per bilinear corner (12 per pixel per pass).
//   pass 0: flow01, shift scale t,     cg = (1-t), cn = (1-t)
//   pass 1: flow10, shift scale (1-t), cg = -t,    cn = t
// ---------------------------------------------------------------------------
__global__ void fwarp_splat_kernel(const float* __restrict__ flow,
                                   const float* __restrict__ tptr,
                                   float* __restrict__ gacc,
                                   float* __restrict__ nrm,
                                   int pass) {
    int i = blockIdx.x * blockDim.x + threadIdx.x;
    if (i >= NHW) return;

    const int n = i / HW;
    const int r = i - n * HW;       // h*W + w
    const int h = r / W_;
    const int w = r - h * W_;

    const float t  = *tptr;
    const float mt = 1.0f - t;
    const float s  = pass ? mt : t;    // shift scale
    const float cg = pass ? -t : mt;   // coeff -> G
    const float cn = pass ?  t : mt;   // coeff -> norm

    const float* f0 = flow + n * (C_ * HW) + r;        // channel 0 plane
    __builtin_prefetch(f0 + HW, 0, 0);                 // global_prefetch_b8 (ch1 plane)
    const float u = f0[0];                             // flo[:,0] -> shifts W (cols)
    const float v = f0[HW];                            // flo[:,1] -> shifts H (rows)

    const float x  = s * v;                            // row displacement
    const float y  = s * u;                            // col displacement
    const float x1 = floorf(x);
    const float y1 = floorf(y);
    const float fx = x - x1;
    const float fy = y - y1;

    const int r0 = h + (int)x1;
    const int c0 = w + (int)y1;

    // Separable Gaussian: exp(-(dx^2+dy^2)) = exp(-dx^2)*exp(-dy^2)
    const float ex[2] = { __expf(-fx * fx), __expf(-(fx - 1.f) * (fx - 1.f)) };
    const float ey[2] = { __expf(-fy * fy), __expf(-(fy - 1.f) * (fy - 1.f)) };

    float* ga = gacc + (size_t)n * (C_ * HW);
    float* nn = nrm  + (size_t)n * HW;

#pragma unroll
    for (int dr = 0; dr < 2; ++dr) {
        const int rr = r0 + dr;
        if (rr < 0 || rr >= H_) continue;
#pragma unroll
        for (int dc = 0; dc < 2; ++dc) {
            const int cc = c0 + dc;
            if (cc < 0 || cc >= W_) continue;
            const float wgt = ex[dr] * ey[dc];
            const int tp = rr * W_ + cc;
            atom_add_f32_dev(ga + tp,      cg * wgt * u);  // G, c=0
            atom_add_f32_dev(ga + HW + tp, cg * wgt * v);  // G, c=1
            atom_add_f32_dev(nn + tp,      cn * wgt);      // normalizer
        }
    }
}

// ---------------------------------------------------------------------------
// Normalize + expand (float4 / B128 path, 4 pixels per thread):
//   sc = (norm>0 ? 1/norm : 1)          (== (1-mask)+mask/denom of reference)
//   flow_t1 = (1-t)*G*sc  (in place over G)
//   flow_t0 =   -t *G*sc  (written, no prior zero needed)
// ---------------------------------------------------------------------------
__global__ void fwarp_norm_kernel(float* __restrict__ out,
                                  const float* __restrict__ nrm,
                                  const float* __restrict__ tptr) {
    constexpr int HW4  = HW / 4;
    constexpr int NHW4 = NHW / 4;
    int g = blockIdx.x * blockDim.x + threadIdx.x;
    if (g >= NHW4) return;

    const float t  = *tptr;
    const float mt = 1.0f - t;

    const int n   = g / HW4;
    const int rem = g - n * HW4;

    const float4 nv = ((const float4*)nrm)[g + 0];  // g indexes (n, pix/4) flat
    float4 sc;
    sc.x = (nv.x > 0.0f) ? (1.0f / nv.x) : 1.0f;
    sc.y = (nv.y > 0.0f) ? (1.0f / nv.y) : 1.0f;
    sc.z = (nv.z > 0.0f) ? (1.0f / nv.z) : 1.0f;
    sc.w = (nv.w > 0.0f) ? (1.0f / nv.w) : 1.0f;

    const size_t b = (size_t)n * (C_ * HW) + (size_t)rem * 4;
    float4* o0c0 = (float4*)(out + b);
    float4* o0c1 = (float4*)(out + b + HW);
    float4* o1c0 = (float4*)(out + NCHW + b);        // G lives here
    float4* o1c1 = (float4*)(out + NCHW + b + HW);

    const float4 g0 = *o1c0;
    const float4 g1 = *o1c1;

    float4 a0, a1, b0, b1;
    a0.x = g0.x * sc.x; a0.y = g0.y * sc.y; a0.z = g0.z * sc.z; a0.w = g0.w * sc.w;
    a1.x = g1.x * sc.x; a1.y = g1.y * sc.y; a1.z = g1.z * sc.z; a1.w = g1.w * sc.w;

    b0.x = -t * a0.x; b0.y = -t * a0.y; b0.z = -t * a0.z; b0.w = -t * a0.w;   // flow_t0 c0
    b1.x = -t * a1.x; b1.y = -t * a1.y; b1.z = -t * a1.z; b1.w = -t * a1.w;   // flow_t0 c1

    a0.x *= mt; a0.y *= mt; a0.z *= mt; a0.w *= mt;                           // flow_t1 c0
    a1.x *= mt; a1.y *= mt; a1.z *= mt; a1.w *= mt;                           // flow_t1 c1

    *o0c0 = b0;
    *o0c1 = b1;
    *o1c0 = a0;
    *o1c1 = a1;
}

// ---------------------------------------------------------------------------
// Launch: zero G-half + norm, splat both flows, normalize/expand.
//   d_in[0] = flow_01 (N,2,H,W) f32
//   d_in[1] = flow_10 (N,2,H,W) f32
//   d_in[2] = t_value (scalar) f32
//   d_out   = [flow_t0 | flow_t1] concatenated (2*NCHW floats)
//   d_ws    = per-pixel normalizer (NHW floats = 29.5 MB)
// ---------------------------------------------------------------------------
extern "C" void kernel_launch(void* const* d_in, const int* in_sizes, int n_in,
                              void* d_out, int out_size, void* d_ws, size_t ws_size,
                              hipStream_t stream) {
    (void)in_sizes; (void)n_in; (void)out_size; (void)ws_size;

    const float* flow01 = (const float*)d_in[0];
    const float* flow10 = (const float*)d_in[1];
    const float* tptr   = (const float*)d_in[2];

    float* out  = (float*)d_out;
    float* gacc = out + NCHW;        // flow_t1 half doubles as G accumulator
    float* nrm  = (float*)d_ws;

    const int TB = 256;  // 8 wave32 waves per block

    const int nz_g = NCHW / 4;       // only the G half needs zeroing
    fwarp_zero_kernel<<<(nz_g + TB - 1) / TB, TB, 0, stream>>>((float4*)gacc, nz_g);
    const int nz_ws = NHW / 4;
    fwarp_zero_kernel<<<(nz_ws + TB - 1) / TB, TB, 0, stream>>>((float4*)d_ws, nz_ws);

    const int gb = (NHW + TB - 1) / TB;
    fwarp_splat_kernel<<<gb, TB, 0, stream>>>(flow01, tptr, gacc, nrm, 0);
    fwarp_splat_kernel<<<gb, TB, 0, stream>>>(flow10, tptr, gacc, nrm, 1);

    const int nb = (NHW / 4 + TB - 1) / TB;
    fwarp_norm_kernel<<<nb, TB, 0, stream>>>(out, nrm, tptr);
}